// GAT_68856915689825
// MI455X (gfx1250) — compile-verified
//
#include <hip/hip_runtime.h>

// GAT layer, N=1024, IN_F=200, H=8, F=16 (OUT_F=128), neg_slope=0.2
// Pipeline:
//   K1: g_l = X@W_l, g_r = X@W_r      via V_WMMA_F32_16X16X4_F32
//   K2: e(i,j,h) -> mask -> softmax_j  fused VALU kernel, writes a (fp32)
//   K3: out = a @ g_r per head         via V_WMMA_F32_16X16X4_F32

#define GAT_N      1024
#define GAT_INF    200
#define GAT_H      8
#define GAT_F      16
#define GAT_OUTF   128
#define GAT_SLOPE  0.2f

typedef __attribute__((ext_vector_type(2))) float v2f;
typedef __attribute__((ext_vector_type(8))) float v8f;

// ---------------------------------------------------------------------------
// K1: one wave per 16x16 output tile. blockIdx = (row_tile, col_tile, which_W)
// A layout (f32 16x4): lane = (half, m); half=0 -> K={0,1}, half=1 -> K={2,3}
// B layout (f32 4x16): mirrored; lane holds B[2*half + {0,1}][m]
// D layout: vgpr r -> row r + 8*half, lane&15 -> col
// ---------------------------------------------------------------------------
__global__ __launch_bounds__(32)
void gat_gemm_wmma(const float* __restrict__ X,
                   const float* __restrict__ Wl,
                   const float* __restrict__ Wr,
                   float* __restrict__ Gl,
                   float* __restrict__ Gr) {
  const int i0   = blockIdx.x * 16;
  const int c0   = blockIdx.y * 16;
  const float* W = blockIdx.z ? Wr : Wl;
  float*       G = blockIdx.z ? Gr : Gl;

  const int lane = threadIdx.x;
  const int m    = lane & 15;
  const int half = lane >> 4;

  const float* xrow = X + (i0 + m) * GAT_INF;

  v8f acc = {};
  for (int k = 0; k < GAT_INF; k += 4) {          // 200 = 50 * K4 steps
    const int ka = k + 2 * half;                  // ka is even -> 8B aligned
    v2f A = *(const v2f*)(xrow + ka);
    v2f B;
    B.x = W[(ka + 0) * GAT_OUTF + c0 + m];
    B.y = W[(ka + 1) * GAT_OUTF + c0 + m];
    acc = __builtin_amdgcn_wmma_f32_16x16x4_f32(
        false, A, false, B, (short)0, acc, false, false);
  }
#pragma unroll
  for (int r = 0; r < 8; ++r)
    G[(i0 + r + 8 * half) * GAT_OUTF + c0 + m] = acc[r];
}

// ---------------------------------------------------------------------------
// K2: fused e + mask + softmax. One 256-thread block per destination row i;
// loops over 8 heads; each thread owns 4 j's. Writes normalized attention
// a[h][i][j] (head-major planes) to workspace.
// e[i,j,h] = sum_f lrelu(g_r[i,h,f] + g_l[j,h,f]) * attn_w[f]
// ---------------------------------------------------------------------------
__global__ __launch_bounds__(256)
void gat_attn_softmax(const float* __restrict__ Gl,
                      const float* __restrict__ Gr,
                      const int*   __restrict__ adj,
                      const float* __restrict__ attn_w,
                      float* __restrict__ Aout) {
  const int i   = blockIdx.x;
  const int tid = threadIdx.x;

  __shared__ float red[256];
  __shared__ float aw[GAT_F];
  __shared__ float gri[GAT_F];

  if (tid < GAT_F) aw[tid] = attn_w[tid];

  for (int h = 0; h < GAT_H; ++h) {
    if (tid < GAT_F) gri[tid] = Gr[i * GAT_OUTF + h * GAT_F + tid];
    __syncthreads();

    float e4[4];
    float p4[4];
    float lmax = -__builtin_inff();
#pragma unroll
    for (int it = 0; it < 4; ++it) {
      const int j = tid + it * 256;
      const float* glj = Gl + j * GAT_OUTF + h * GAT_F;
      float s = 0.f;
#pragma unroll
      for (int f = 0; f < GAT_F; ++f) {
        float v = gri[f] + glj[f];
        v = (v > 0.f) ? v : GAT_SLOPE * v;
        s = fmaf(v, aw[f], s);
      }
      e4[it] = adj[i * GAT_N + j] ? s : -__builtin_inff();
      lmax = fmaxf(lmax, e4[it]);
    }

    // block-wide max
    red[tid] = lmax;
    __syncthreads();
    for (int off = 128; off > 0; off >>= 1) {
      if (tid < off) red[tid] = fmaxf(red[tid], red[tid + off]);
      __syncthreads();
    }
    const float M = red[0];
    __syncthreads();

    float lsum = 0.f;
#pragma unroll
    for (int it = 0; it < 4; ++it) {
      p4[it] = __expf(e4[it] - M);   // masked: exp(-inf) = 0
      lsum += p4[it];
    }

    // block-wide sum
    red[tid] = lsum;
    __syncthreads();
    for (int off = 128; off > 0; off >>= 1) {
      if (tid < off) red[tid] += red[tid + off];
      __syncthreads();
    }
    const float invS = 1.f / red[0];
    __syncthreads();

    float* arow = Aout + ((size_t)h * GAT_N + i) * GAT_N;
#pragma unroll
    for (int it = 0; it < 4; ++it)
      arow[tid + it * 256] = p4[it] * invS;

    __syncthreads();   // protect gri/red before next head
  }
}

// ---------------------------------------------------------------------------
// K3: out[i, h*16+f] = sum_j a[h][i][j] * g_r[j, h*16+f]
// One wave per (16-row tile, head): 16x16 output tile, K=1024 in steps of 4.
// ---------------------------------------------------------------------------
__global__ __launch_bounds__(32)
void gat_aggregate_wmma(const float* __restrict__ Aplanes,
                        const float* __restrict__ Gr,
                        float* __restrict__ Out) {
  const int i0 = blockIdx.x * 16;
  const int h  = blockIdx.y;

  const int lane = threadIdx.x;
  const int m    = lane & 15;
  const int half = lane >> 4;

  const float* Ap = Aplanes + (size_t)h * GAT_N * GAT_N + (size_t)(i0 + m) * GAT_N;
  const float* Bp = Gr + h * GAT_F + m;

  v8f acc = {};
  for (int j0 = 0; j0 < GAT_N; j0 += 4) {         // 256 K4 steps
    const int ka = j0 + 2 * half;                 // even -> 8B aligned
    v2f A = *(const v2f*)(Ap + ka);
    v2f B;
    B.x = Bp[(ka + 0) * GAT_OUTF];
    B.y = Bp[(ka + 1) * GAT_OUTF];
    acc = __builtin_amdgcn_wmma_f32_16x16x4_f32(
        false, A, false, B, (short)0, acc, false, false);
  }
#pragma unroll
  for (int r = 0; r < 8; ++r)
    Out[(i0 + r + 8 * half) * GAT_OUTF + h * GAT_F + m] = acc[r];
}

// ---------------------------------------------------------------------------
// Host launcher. Inputs: h[1,1024,200] f32, adj[1,1024,1024] i32,
// W_l[200,128] f32, W_r[200,128] f32, attn_w[16] f32. Out: [1,1024,128] f32.
// Workspace: g_l (512KB) | g_r (512KB) | a planes (32MB) = ~33MB.
// ---------------------------------------------------------------------------
extern "C" void kernel_launch(void* const* d_in, const int* in_sizes, int n_in,
                              void* d_out, int out_size, void* d_ws, size_t ws_size,
                              hipStream_t stream) {
  (void)in_sizes; (void)n_in; (void)out_size; (void)ws_size;

  const float* hX  = (const float*)d_in[0];
  const int*   adj = (const int*)  d_in[1];
  const float* Wl  = (const float*)d_in[2];
  const float* Wr  = (const float*)d_in[3];
  const float* aw  = (const float*)d_in[4];
  float* out = (float*)d_out;

  float* gl = (float*)d_ws;
  float* gr = gl + (size_t)GAT_N * GAT_OUTF;
  float* ap = gr + (size_t)GAT_N * GAT_OUTF;     // 8 * 1024 * 1024 floats

  gat_gemm_wmma<<<dim3(GAT_N / 16, GAT_OUTF / 16, 2), 32, 0, stream>>>(
      hX, Wl, Wr, gl, gr);
  gat_attn_softmax<<<dim3(GAT_N), 256, 0, stream>>>(gl, gr, adj, aw, ap);
  gat_aggregate_wmma<<<dim3(GAT_N / 16, GAT_H), 32, 0, stream>>>(ap, gr, out);
}